// MolGAT_9174050144473
// MI455X (gfx1250) — compile-verified
//
#include <hip/hip_runtime.h>
#include <cstddef>
#include <cstdint>

static constexpr int TNF  = 24;   // node input features
static constexpr int TEF  = 6;    // edge input features
static constexpr int THID = 64;   // per-head hidden
static constexpr int TEMB = 256;  // output embedding

typedef float v2f __attribute__((ext_vector_type(2)));
typedef float v8f __attribute__((ext_vector_type(8)));

// Monotone order-preserving float<->uint map so atomicMax(uint) == float max.
__device__ __forceinline__ unsigned int encode_ord(float f) {
  unsigned int u = __float_as_uint(f);
  return (u & 0x80000000u) ? ~u : (u | 0x80000000u);
}
__device__ __forceinline__ float decode_ord(unsigned int u) {
  return __uint_as_float((u & 0x80000000u) ? (u & 0x7fffffffu) : ~u);
}

// ============================================================================
// WMMA f32 GEMM:  C[M,Kout] = act(A[M,Kin] * W[Kout,Kin]^T + bias)
// One wave computes one 16x16 output tile via V_WMMA_F32_16X16X4_F32.
// Requires M%16==0, Kout%16==0, Kin%4==0 (all true for this problem).
// A-frag layout (ISA 7.12.2, 32-bit A 16x4): lanes0-15 hold K=0/1, lanes16-31 K=2/3.
// act: 0=none, 1=elu, 2=relu
// ============================================================================
__global__ __launch_bounds__(256) void gemm_f32_wmma(
    const float* __restrict__ A, const float* __restrict__ W,
    const float* __restrict__ bias, float* __restrict__ C,
    int M, int Kin, int Kout, int act)
{
  const int wave = threadIdx.x >> 5;
  const int lane = threadIdx.x & 31;
  const int tilesX = Kout >> 4;
  const int totalTiles = (M >> 4) * tilesX;
  const int tile = blockIdx.x * 8 + wave;
  if (tile >= totalTiles) return;              // whole-wave uniform: EXEC stays all-1s
  const int tm = (tile / tilesX) << 4;
  const int to = (tile % tilesX) << 4;
  const int half = lane >> 4;                  // 0: K lanes {0,1}, 1: K lanes {2,3}
  const int l16  = lane & 15;

  const float* arow = A + (size_t)(tm + l16) * Kin + 2 * half;
  const float* wrow = W + (size_t)(to + l16) * Kin + 2 * half;  // B[k][o] = W[o][k]

  v8f acc = {0.f, 0.f, 0.f, 0.f, 0.f, 0.f, 0.f, 0.f};
  for (int kk = 0; kk < Kin; kk += 4) {
    v2f a, b;
    a[0] = arow[kk];  a[1] = arow[kk + 1];
    b[0] = wrow[kk];  b[1] = wrow[kk + 1];
    acc = __builtin_amdgcn_wmma_f32_16x16x4_f32(false, a, false, b, (short)0, acc,
                                                false, false);
  }
  const int o = to + l16;
  const float bv = bias ? bias[o] : 0.0f;
#pragma unroll
  for (int r = 0; r < 8; ++r) {
    const int m = tm + r + 8 * half;           // C layout: vgpr r -> row r (+8 for hi lanes)
    float v = acc[r] + bv;
    if (act == 1)      v = (v > 0.f) ? v : (expf(v) - 1.f);   // elu
    else if (act == 2) v = fmaxf(v, 0.f);                     // relu
    C[(size_t)m * Kout + o] = v;
  }
}

// ---------------------------------------------------------------------------
// Self-loop edge features: in-degree count + segment-sum of edge_attr over dst
// ---------------------------------------------------------------------------
__global__ void degree_edgesum_kernel(const float* __restrict__ ea,
                                      const int* __restrict__ dst, int E,
                                      float* __restrict__ cnt, float* __restrict__ esum)
{
  int e = blockIdx.x * blockDim.x + threadIdx.x;
  if (e >= E) return;
  int t = dst[e];
  atomicAdd(&cnt[t], 1.0f);
  for (int f = 0; f < TEF; ++f)
    atomicAdd(&esum[(size_t)t * TEF + f], ea[(size_t)e * TEF + f]);
}

__global__ void eself_div_kernel(const float* __restrict__ cnt, float* __restrict__ esum, int Nn)
{
  int i = blockIdx.x * blockDim.x + threadIdx.x;
  if (i >= Nn * TEF) return;
  esum[i] = esum[i] / fmaxf(cnt[i / TEF], 1.0f);
}

// ---------------------------------------------------------------------------
// alpha_src[n,h] = <xh[n,h,:], as[h,:]>,  alpha_dst[n,h] = <xh[n,h,:], ad[h,:]>
// ---------------------------------------------------------------------------
__global__ void alpha_nodes_kernel(const float* __restrict__ xh,
                                   const float* __restrict__ as_w,
                                   const float* __restrict__ ad_w,
                                   int Nn, int H,
                                   float* __restrict__ asrc, float* __restrict__ adst)
{
  int i = blockIdx.x * blockDim.x + threadIdx.x;
  if (i >= Nn * H) return;
  int n = i / H, h = i % H;
  const float* xp = xh + (size_t)n * H * THID + h * THID;
  const float* sp = as_w + h * THID;
  const float* dp = ad_w + h * THID;
  float s = 0.f, d = 0.f;
  for (int k = 0; k < THID; ++k) { float v = xp[k]; s += v * sp[k]; d += v * dp[k]; }
  asrc[i] = s; adst[i] = d;
}

// va[h,f] = sum_d We[h*64+d, f] * ae[h, d]   (folds the E'x(H*64) edge GEMM away)
__global__ void edge_attn_vec_kernel(const float* __restrict__ We,
                                     const float* __restrict__ ae_w,
                                     int H, float* __restrict__ va)
{
  int i = threadIdx.x;
  if (i >= H * TEF) return;
  int h = i / TEF, f = i % TEF;
  float s = 0.f;
  for (int d = 0; d < THID; ++d)
    s += We[(size_t)(h * THID + d) * TEF + f] * ae_w[h * THID + d];
  va[i] = s;
}

// logits + segment max (edges 0..E-1 real, E..E+N-1 self loops)
__global__ void logits_kernel(const float* __restrict__ edge_attr,
                              const float* __restrict__ eself,
                              const float* __restrict__ va,
                              const float* __restrict__ asrc,
                              const float* __restrict__ adst,
                              const int* __restrict__ src, const int* __restrict__ dst,
                              int E, int Nn, int H,
                              float* __restrict__ logit, unsigned int* __restrict__ mbuf)
{
  int e = blockIdx.x * blockDim.x + threadIdx.x;
  if (e >= E + Nn) return;
  const float* ea = (e < E) ? (edge_attr + (size_t)e * TEF) : (eself + (size_t)(e - E) * TEF);
  int s = (e < E) ? src[e] : (e - E);
  int t = (e < E) ? dst[e] : (e - E);
  float eav[TEF];
  for (int f = 0; f < TEF; ++f) eav[f] = ea[f];
  for (int h = 0; h < H; ++h) {
    float z = asrc[(size_t)s * H + h] + adst[(size_t)t * H + h];
    for (int f = 0; f < TEF; ++f) z += eav[f] * va[h * TEF + f];
    z = (z > 0.f) ? z : 0.2f * z;             // leaky_relu(0.2)
    logit[(size_t)e * H + h] = z;
    atomicMax(&mbuf[(size_t)t * H + h], encode_ord(z));
  }
}

// ex = exp(logit - m[dst]); denom[dst] += ex  (in-place on logit buffer)
__global__ void expsum_kernel(float* __restrict__ exv,
                              const unsigned int* __restrict__ mbuf,
                              const int* __restrict__ dst, int E, int Nn, int H,
                              float* __restrict__ denom)
{
  long long i = (long long)blockIdx.x * blockDim.x + threadIdx.x;
  long long total = (long long)(E + Nn) * H;
  if (i >= total) return;
  int e = (int)(i / H), h = (int)(i % H);
  int t = (e < E) ? dst[e] : (e - E);
  float m = decode_ord(mbuf[(size_t)t * H + h]);
  float x = expf(exv[i] - m);
  exv[i] = x;
  atomicAdd(&denom[(size_t)t * H + h], x);
}

// agg[dst,h,:] += xh[src,h,:] * att ; 8 floats per thread
__global__ void aggregate_kernel(const float* __restrict__ xh,
                                 const float* __restrict__ exv,
                                 const float* __restrict__ denom,
                                 const int* __restrict__ src, const int* __restrict__ dst,
                                 int E, int Nn, int H, float* __restrict__ agg)
{
  const int HD  = H * THID;
  const int tpe = HD >> 3;                    // threads per edge
  long long gid = (long long)blockIdx.x * blockDim.x + threadIdx.x;
  long long total = (long long)(E + Nn) * tpe;
  if (gid >= total) return;
  int e = (int)(gid / tpe);
  int part = (int)(gid % tpe);
  int d0 = part << 3;
  int h = d0 >> 6;                            // 8 | 64, so head constant in chunk
  int s = (e < E) ? src[e] : (e - E);
  int t = (e < E) ? dst[e] : (e - E);
  float att = exv[(size_t)e * H + h] / (denom[(size_t)t * H + h] + 1e-16f);
  const float* xs = xh + (size_t)s * HD + d0;
  float* ag = agg + (size_t)t * HD + d0;
#pragma unroll
  for (int j = 0; j < 8; ++j) atomicAdd(&ag[j], xs[j] * att);
}

// out = bn(elu(agg + b)) (+ residual)
__global__ void post_bn_res_kernel(const float* __restrict__ agg, const float* __restrict__ gb,
                                   const float* __restrict__ bng, const float* __restrict__ bnb,
                                   const float* __restrict__ bnm, const float* __restrict__ bnv,
                                   const float* __restrict__ hprev, float* __restrict__ out,
                                   int Nn, int outd)
{
  long long i = (long long)blockIdx.x * blockDim.x + threadIdx.x;
  if (i >= (long long)Nn * outd) return;
  int o = (int)(i % outd);
  float z = agg[i] + gb[o];
  z = (z > 0.f) ? z : (expf(z) - 1.f);
  z = (z - bnm[o]) * rsqrtf(bnv[o] + 1e-5f) * bng[o] + bnb[o];
  if (hprev) z += hprev[i];
  out[i] = z;
}

// ---------------------------------------------------------------------------
// Pooling
// ---------------------------------------------------------------------------
__global__ void pool_count_kernel(const int* __restrict__ batch, int Nn, float* __restrict__ gc)
{
  int n = blockIdx.x * blockDim.x + threadIdx.x;
  if (n >= Nn) return;
  atomicAdd(&gc[batch[n]], 1.0f);
}

__global__ void pool_feat_kernel(const float* __restrict__ h3, const int* __restrict__ batch,
                                 int Nn, float* __restrict__ psum, unsigned int* __restrict__ pmax)
{
  long long i = (long long)blockIdx.x * blockDim.x + threadIdx.x;
  if (i >= (long long)Nn * THID) return;
  int n = (int)(i / THID), d = (int)(i % THID);
  int g = batch[n];
  float v = h3[i];
  atomicAdd(&psum[(size_t)g * THID + d], v);
  atomicMax(&pmax[(size_t)g * THID + d], encode_ord(v));
}

__global__ void pooled_build_kernel(const float* __restrict__ psum,
                                    const unsigned int* __restrict__ pmax,
                                    const float* __restrict__ gc, int G,
                                    float* __restrict__ pooled)
{
  int i = blockIdx.x * blockDim.x + threadIdx.x;
  if (i >= G * 2 * THID) return;
  int g = i / (2 * THID), k = i % (2 * THID);
  float v;
  if (k < THID) v = psum[(size_t)g * THID + k] / fmaxf(gc[g], 1.f);
  else          v = (gc[g] > 0.f) ? decode_ord(pmax[(size_t)g * THID + (k - THID)]) : 0.f;
  pooled[i] = v;
}

// ============================================================================
extern "C" void kernel_launch(void* const* d_in, const int* in_sizes, int n_in,
                              void* d_out, int out_size, void* d_ws, size_t ws_size,
                              hipStream_t stream)
{
  (void)n_in; (void)ws_size;
  // setup_inputs() insertion order: x, edge_attr, params(34 leaves in insertion
  // order), edge_index, batch
  int idx = 0;
  const float* x         = (const float*)d_in[idx++];   // [N,24]
  const float* edge_attr = (const float*)d_in[idx++];   // [E,6]
  const float* W_in = (const float*)d_in[idx++];        // [256,24]
  const float* b_in = (const float*)d_in[idx++];        // [256]
  struct GatParams { const float *W, *We, *as, *ad, *ae, *b, *g, *bb, *m, *v; };
  GatParams gp[3];
  for (int l = 0; l < 3; ++l) {
    gp[l].W  = (const float*)d_in[idx++];
    gp[l].We = (const float*)d_in[idx++];
    gp[l].as = (const float*)d_in[idx++];
    gp[l].ad = (const float*)d_in[idx++];
    gp[l].ae = (const float*)d_in[idx++];
    gp[l].b  = (const float*)d_in[idx++];
    gp[l].g  = (const float*)d_in[idx++];   // bn gamma
    gp[l].bb = (const float*)d_in[idx++];   // bn beta
    gp[l].m  = (const float*)d_in[idx++];   // bn mean
    gp[l].v  = (const float*)d_in[idx++];   // bn var
  }
  const float* W_proj = (const float*)d_in[idx++];      // [256,128]
  const float* b_proj = (const float*)d_in[idx++];      // [256]
  const int* edge_index = (const int*)d_in[idx++];      // [2,E] int32
  const int* batch      = (const int*)d_in[idx++];      // [N]   int32

  const int N = in_sizes[0] / TNF;
  const int E = in_sizes[1] / TEF;
  const int G = out_size / TEMB;
  const int* src = edge_index;
  const int* dst = edge_index + E;
  const int WID = 4 * THID;                 // 256 = HEADS*HID

  // ---- workspace carve (256B-aligned regions) ----
  char* wsp = (char*)d_ws;
  auto carve = [&](size_t bytes) -> void* {
    void* p = (void*)wsp;
    wsp += (bytes + 255) & ~(size_t)255;
    return p;
  };
  float* cnt    = (float*)carve((size_t)N * 4);
  float* eself  = (float*)carve((size_t)N * TEF * 4);
  float* bufA   = (float*)carve((size_t)N * WID * 4);   // h0, later h2
  float* bufB   = (float*)carve((size_t)N * WID * 4);   // h1
  float* xh     = (float*)carve((size_t)N * WID * 4);
  float* agg    = (float*)carve((size_t)N * WID * 4);
  float* asrc   = (float*)carve((size_t)N * 4 * 4);
  float* adst   = (float*)carve((size_t)N * 4 * 4);
  unsigned int* mbuf = (unsigned int*)carve((size_t)N * 4 * 4);
  float* denom  = (float*)carve((size_t)N * 4 * 4);
  float* exbuf  = (float*)carve((size_t)(E + N) * 4 * 4);
  float* va     = (float*)carve(4 * TEF * 4);
  float* h3     = (float*)carve((size_t)N * THID * 4);
  float* gc     = (float*)carve((size_t)G * 4);
  float* psum   = (float*)carve((size_t)G * THID * 4);
  unsigned int* pmax = (unsigned int*)carve((size_t)G * THID * 4);
  float* pooled = (float*)carve((size_t)G * 2 * THID * 4);

  auto blocks1d = [](long long n) { return (unsigned)((n + 255) / 256); };
  auto launch_gemm = [&](const float* A, const float* W, const float* bias, float* C,
                         int M, int Kin, int Kout, int act) {
    int tiles = (M / 16) * (Kout / 16);
    int blocks = (tiles + 7) / 8;             // 8 waves (16x16 tiles) per 256-thread block
    hipLaunchKernelGGL(gemm_f32_wmma, dim3(blocks), dim3(256), 0, stream,
                       A, W, bias, C, M, Kin, Kout, act);
  };

  // ---- self-loop edge features (mean of incoming edge feats) ----
  hipMemsetAsync(cnt, 0, (size_t)N * 4, stream);
  hipMemsetAsync(eself, 0, (size_t)N * TEF * 4, stream);
  hipLaunchKernelGGL(degree_edgesum_kernel, dim3(blocks1d(E)), dim3(256), 0, stream,
                     edge_attr, dst, E, cnt, eself);
  hipLaunchKernelGGL(eself_div_kernel, dim3(blocks1d((long long)N * TEF)), dim3(256), 0, stream,
                     cnt, eself, N);

  // ---- h0 = elu(x @ W_in^T + b_in)  [WMMA f32] ----
  launch_gemm(x, W_in, b_in, bufA, N, TNF, WID, /*elu*/1);

  // ---- three GAT layers ----
  const float* hin[3]  = { bufA, bufB, bufA };
  float*       hout[3] = { bufB, bufA, h3  };
  const float* hres[3] = { bufA, bufB, nullptr };
  const int    Hs[3]   = { 4, 4, 1 };

  for (int l = 0; l < 3; ++l) {
    const int H = Hs[l], HD = H * THID;
    // xh = h @ W^T  [WMMA f32]
    launch_gemm(hin[l], gp[l].W, nullptr, xh, N, WID, HD, /*none*/0);
    hipLaunchKernelGGL(alpha_nodes_kernel, dim3(blocks1d((long long)N * H)), dim3(256), 0, stream,
                       xh, gp[l].as, gp[l].ad, N, H, asrc, adst);
    hipLaunchKernelGGL(edge_attn_vec_kernel, dim3(1), dim3(32), 0, stream,
                       gp[l].We, gp[l].ae, H, va);
    hipMemsetAsync(mbuf, 0, (size_t)N * H * 4, stream);
    hipMemsetAsync(denom, 0, (size_t)N * H * 4, stream);
    hipMemsetAsync(agg, 0, (size_t)N * HD * 4, stream);
    hipLaunchKernelGGL(logits_kernel, dim3(blocks1d((long long)E + N)), dim3(256), 0, stream,
                       edge_attr, eself, va, asrc, adst, src, dst, E, N, H, exbuf, mbuf);
    hipLaunchKernelGGL(expsum_kernel, dim3(blocks1d((long long)(E + N) * H)), dim3(256), 0, stream,
                       exbuf, mbuf, dst, E, N, H, denom);
    hipLaunchKernelGGL(aggregate_kernel, dim3(blocks1d((long long)(E + N) * (HD / 8))),
                       dim3(256), 0, stream, xh, exbuf, denom, src, dst, E, N, H, agg);
    hipLaunchKernelGGL(post_bn_res_kernel, dim3(blocks1d((long long)N * HD)), dim3(256), 0, stream,
                       agg, gp[l].b, gp[l].g, gp[l].bb, gp[l].m, gp[l].v, hres[l], hout[l], N, HD);
  }

  // ---- pooling (mean + max per molecule) ----
  hipMemsetAsync(gc, 0, (size_t)G * 4, stream);
  hipMemsetAsync(psum, 0, (size_t)G * THID * 4, stream);
  hipMemsetAsync(pmax, 0, (size_t)G * THID * 4, stream);
  hipLaunchKernelGGL(pool_count_kernel, dim3(blocks1d(N)), dim3(256), 0, stream, batch, N, gc);
  hipLaunchKernelGGL(pool_feat_kernel, dim3(blocks1d((long long)N * THID)), dim3(256), 0, stream,
                     h3, batch, N, psum, pmax);
  hipLaunchKernelGGL(pooled_build_kernel, dim3(blocks1d((long long)G * 2 * THID)), dim3(256), 0,
                     stream, psum, pmax, gc, G, pooled);

  // ---- emb = relu(pooled @ W_proj^T + b_proj)  [WMMA f32] ----
  launch_gemm(pooled, W_proj, b_proj, (float*)d_out, G, 2 * THID, TEMB, /*relu*/2);
}